// GroupAttention_M_8770323218747
// MI455X (gfx1250) — compile-verified
//
#include <hip/hip_runtime.h>
#include <math.h>

typedef float v2f __attribute__((ext_vector_type(2)));
typedef float v4f __attribute__((ext_vector_type(4)));
typedef float v8f __attribute__((ext_vector_type(8)));

#define NPAIR 16384   // SEQ*NENV
#define NG    12
#define GD    128
#define RD    256
// ws layout (floats): [0..127] v = Wg@Wa2, [128..383] u = Wr@Wa1, [384] c = br.Wa1+bg.Wa2+ba

__global__ __launch_bounds__(512) void ga_precompute(
    const float* __restrict__ Wr, const float* __restrict__ br,
    const float* __restrict__ Wg, const float* __restrict__ bg,
    const float* __restrict__ Wa, const float* __restrict__ ba,
    float* __restrict__ ws)
{
  int t = threadIdx.x;
  if (t < 128) {
    const float* row = Wg + t * 128;
    float s = 0.f;
    #pragma unroll 4
    for (int j = 0; j < 128; j += 4) {
      v4f a = *(const v4f*)(row + j);
      v4f b = *(const v4f*)(Wa + 128 + j);
      s += a[0]*b[0] + a[1]*b[1] + a[2]*b[2] + a[3]*b[3];
    }
    ws[t] = s;
  } else if (t < 384) {
    int k = t - 128;
    const float* row = Wr + k * 128;
    float s = 0.f;
    #pragma unroll 4
    for (int j = 0; j < 128; j += 4) {
      v4f a = *(const v4f*)(row + j);
      v4f b = *(const v4f*)(Wa + j);
      s += a[0]*b[0] + a[1]*b[1] + a[2]*b[2] + a[3]*b[3];
    }
    ws[128 + k] = s;
  } else if (t == 384) {
    float s = ba[0];
    for (int j = 0; j < 128; ++j) s += br[j]*Wa[j] + bg[j]*Wa[128 + j];
    ws[384] = s;
  }
}

__global__ __launch_bounds__(256) void ga_main(
    const float* __restrict__ rs, const float* __restrict__ ge,
    const float* __restrict__ ws, float* __restrict__ out_w,
    float* __restrict__ out_wt)
{
  const int lane = threadIdx.x & 31;
  const int pair = (blockIdx.x << 3) + (threadIdx.x >> 5);
  const int n    = lane & 15;        // group index served by this lane
  const int h    = lane >> 4;        // K half (A: VGPR0/1 = K{2h,2h+1})
  const int grow = (n < NG) ? n : (NG - 1);   // clamp pad rows to valid data

  const float* gep = ge + (size_t)pair * (NG * GD);
  const float* gb  = gep + grow * GD + 2 * h;   // B operand: lane=N(group), VGPR=K within half
  const float* va  = ws + 2 * h;                // A operand: v broadcast over all M rows

  // ---- scores[g] = v . GE[g,:] via 32 chained V_WMMA_F32_16X16X4_F32 ----
  v8f acc = {0.f,0.f,0.f,0.f,0.f,0.f,0.f,0.f};
  #pragma unroll
  for (int t = 0; t < 32; ++t) {
    v2f a = *(const v2f*)(va + 4 * t);
    v2f b = *(const v2f*)(gb + 4 * t);
    acc = __builtin_amdgcn_wmma_f32_16x16x4_f32(
        /*neg_a=*/false, a, /*neg_b=*/false, b,
        /*c_mod=*/(short)0, acc, /*reuse_a=*/false, /*reuse_b=*/false);
  }
  // D rows all identical (A rows broadcast): lane l holds score[l&15] in VGPR0
  float score = acc[0];

  // ---- robot score (constant over g, kept for reference-faithful weights) ----
  const float* rp = rs + (size_t)pair * RD + lane * 8;
  const float* up = ws + 128 + lane * 8;
  v4f r0 = *(const v4f*)rp;       v4f r1 = *(const v4f*)(rp + 4);
  v4f u0 = *(const v4f*)up;       v4f u1 = *(const v4f*)(up + 4);
  float rsum = r0[0]*u0[0] + r0[1]*u0[1] + r0[2]*u0[2] + r0[3]*u0[3]
             + r1[0]*u1[0] + r1[1]*u1[1] + r1[2]*u1[2] + r1[3]*u1[3];
  #pragma unroll
  for (int m = 16; m >= 1; m >>= 1) rsum += __shfl_xor(rsum, m, 32);
  score += rsum + ws[384];

  // ---- softmax over 12 groups (within each 16-lane half; halves identical) ----
  float sm = (n < NG) ? score : -INFINITY;
  #pragma unroll
  for (int m = 8; m >= 1; m >>= 1) sm = fmaxf(sm, __shfl_xor(sm, m, 16));
  float e  = (n < NG) ? __expf(score - sm) : 0.f;
  float es = e;
  #pragma unroll
  for (int m = 8; m >= 1; m >>= 1) es += __shfl_xor(es, m, 16);
  float w = e / es;
  if (lane < NG) out_wt[pair * NG + lane] = w;

  // ---- weighted sum of raw group embeddings (coalesced b128, hits WGP$) ----
  v4f accw = {0.f, 0.f, 0.f, 0.f};
  #pragma unroll
  for (int g = 0; g < NG; ++g) {
    float wg = __shfl(w, g, 32);
    v4f gv = *(const v4f*)(gep + g * GD + lane * 4);
    accw[0] += wg * gv[0]; accw[1] += wg * gv[1];
    accw[2] += wg * gv[2]; accw[3] += wg * gv[3];
  }
  *(v4f*)(out_w + (size_t)pair * GD + lane * 4) = accw;
}

extern "C" void kernel_launch(void* const* d_in, const int* in_sizes, int n_in,
                              void* d_out, int out_size, void* d_ws, size_t ws_size,
                              hipStream_t stream) {
  const float* rs = (const float*)d_in[0];   // robot_states   (S,E,1,256)
  const float* ge = (const float*)d_in[1];   // group_embeddings (S,E,12,128)
  const float* Wr = (const float*)d_in[2];
  const float* br = (const float*)d_in[3];
  const float* Wg = (const float*)d_in[4];
  const float* bg = (const float*)d_in[5];
  const float* Wa = (const float*)d_in[6];
  const float* ba = (const float*)d_in[7];
  float* ws  = (float*)d_ws;
  float* out = (float*)d_out;

  ga_precompute<<<1, 512, 0, stream>>>(Wr, br, Wg, bg, Wa, ba, ws);
  ga_main<<<NPAIR / 8, 256, 0, stream>>>(rs, ge, ws, out, out + (size_t)NPAIR * GD);
}